// EdgeguidedNormalRankingLoss_163208757556
// MI455X (gfx1250) — compile-verified
//
#include <hip/hip_runtime.h>
#include <math.h>

#ifndef __has_builtin
#define __has_builtin(x) 0
#endif

#if __has_builtin(__builtin_amdgcn_global_load_async_to_lds_b32)
#define HAVE_ASYNC_LDS 1
#endif

#define H_IMG 256
#define W_IMG 832
#define S_PAIRS 10000
#define TILE_W 32
#define TILE_H 8
#define HALO_W (TILE_W + 2)
#define HALO_H (TILE_H + 2)
#define HALO_N (HALO_W * HALO_H)

typedef __attribute__((ext_vector_type(2))) float v2f;
typedef __attribute__((ext_vector_type(8))) float v8f;

// ---------- helpers ----------

__device__ inline void atomic_max_pos(float* addr, float v) {
  // valid for non-negative floats with 0-initialized destination
  atomicMax((int*)addr, __float_as_int(v));
}

__device__ inline unsigned pcg_hash(unsigned x) {
  x = x * 747796405u + 2891336453u;
  unsigned w = ((x >> ((x >> 28u) + 4u)) ^ x) * 277803737u;
  return (w >> 22u) ^ w;
}

__device__ inline void async_copy_f32(const float* g, float* l) {
#ifdef HAVE_ASYNC_LDS
  __builtin_amdgcn_global_load_async_to_lds_b32(
      (__attribute__((address_space(1))) int*)(g),
      (__attribute__((address_space(3))) int*)(l), 0, 0);
#else
  *l = *g;
#endif
}

__device__ inline void async_wait_all() {
#ifdef HAVE_ASYNC_LDS
#if __has_builtin(__builtin_amdgcn_s_wait_asynccnt)
  __builtin_amdgcn_s_wait_asynccnt(0);
#else
  asm volatile("s_wait_asynccnt 0" ::: "memory");
#endif
#endif
}

// ---------- K0: zero the small-scalar region of workspace ----------

__global__ void k_init(float* smalls) {
  if (threadIdx.x < 64) smalls[threadIdx.x] = 0.0f;
}

// ---------- K1: fused Sobel edge/theta maps, LDS-tiled w/ async staging ----------
// mode 0: gt_depths (c=1)  -> e_dep (+ global max of e_dep)
// mode 1: images ch0 (c=1) -> e_img, t_img
// mode 2: targets_normal (c=3, abs-mean grouped) -> e_nrm, t_nrm

__global__ void k_edges(const float* __restrict__ gt, const float* __restrict__ img,
                        const float* __restrict__ nrm,
                        float* e_img, float* t_img, float* e_nrm, float* t_nrm,
                        float* e_dep, float* gmax, int n, int h, int w) {
  __shared__ float tile[3 * HALO_N];
  const int mode = blockIdx.z;
  const int rbpi = (h + TILE_H - 1) / TILE_H;
  const int im = blockIdx.y / rbpi;
  const int by = blockIdx.y % rbpi;
  const int iy = by * TILE_H + threadIdx.y;
  const int ix = blockIdx.x * TILE_W + threadIdx.x;
  const size_t hw = (size_t)h * w;
  const int gy0 = by * TILE_H - 1;
  const int gx0 = blockIdx.x * TILE_W - 1;
  const int cc = (mode == 2) ? 3 : 1;
  const float* base = (mode == 0) ? (gt + (size_t)im * hw)
                    : (mode == 1) ? (img + (size_t)im * 3 * hw)
                                  : (nrm + (size_t)im * 3 * hw);
  const int tid = threadIdx.y * TILE_W + threadIdx.x;

  // cooperative halo staging into LDS (async-to-LDS on CDNA5)
  for (int e = tid; e < cc * HALO_N; e += TILE_W * TILE_H) {
    int c = e / HALO_N;
    int r = (e % HALO_N) / HALO_W;
    int q = (e % HALO_N) % HALO_W;
    int gy = gy0 + r; gy = gy < 0 ? 0 : (gy > h - 1 ? h - 1 : gy);
    int gx = gx0 + q; gx = gx < 0 ? 0 : (gx > w - 1 ? w - 1 : gx);
    async_copy_f32(base + (size_t)c * hw + (size_t)gy * w + gx,
                   &tile[c * HALO_N + r * HALO_W + q]);
  }
  async_wait_all();
  __syncthreads();

  const int ly = threadIdx.y + 1, lx = threadIdx.x + 1;
  const bool inimg = (iy < h) && (ix < w);
  const bool interior = inimg && (iy >= 1) && (iy <= h - 2) && (ix >= 1) && (ix <= w - 2);
  float gxa = 0.f, gya = 0.f;
  for (int c = 0; c < cc; ++c) {
    const float* t = &tile[c * HALO_N];
    float p00 = t[(ly - 1) * HALO_W + lx - 1], p01 = t[(ly - 1) * HALO_W + lx],
          p02 = t[(ly - 1) * HALO_W + lx + 1];
    float p10 = t[ly * HALO_W + lx - 1], p12 = t[ly * HALO_W + lx + 1];
    float p20 = t[(ly + 1) * HALO_W + lx - 1], p21 = t[(ly + 1) * HALO_W + lx],
          p22 = t[(ly + 1) * HALO_W + lx + 1];
    float gx = (p02 - p00) + 2.f * (p12 - p10) + (p22 - p20);
    float gy = (p00 + 2.f * p01 + p02) - (p20 + 2.f * p21 + p22);
    if (mode == 2) { gxa += fabsf(gx); gya += fabsf(gy); }
    else { gxa = gx; gya = gy; }
  }
  if (mode == 2) { gxa *= (1.f / 3.f); gya *= (1.f / 3.f); }
  float e_out = 0.f, th_out = 0.f;
  if (interior) {
    e_out = sqrtf(gxa * gxa + gya * gya);
    th_out = atan2f(gya, gxa);
  }
  if (inimg) {
    size_t p = (size_t)im * hw + (size_t)iy * w + ix;
    if (mode == 0) e_dep[p] = e_out;
    else if (mode == 1) { e_img[p] = e_out; t_img[p] = th_out; }
    else { e_nrm[p] = e_out; t_nrm[p] = th_out; }
  }
  if (mode == 0) {
    float v = inimg ? e_out : 0.f;
    for (int off = 16; off; off >>= 1) v = fmaxf(v, __shfl_xor(v, off, 32));
    if ((tid & 31) == 0) atomic_max_pos(gmax, v);
  }
}

// ---------- K2: depth-edge dilation suppression + border-5 zero + per-image maxes ----------

__global__ void k_mask(float* e_img, float* e_nrm, const float* __restrict__ e_dep,
                       const float* __restrict__ gmax, float* maxes, int n, int h, int w) {
  const size_t hw = (size_t)h * w;
  const size_t p = (size_t)blockIdx.x * blockDim.x + threadIdx.x;
  const int im = (int)(p / hw);
  const size_t r = p % hw;
  const int y = (int)(r / w), x = (int)(r % w);
  const float thr = 0.1f * gmax[0];
  const float* ed = e_dep + (size_t)im * hw;
  bool dil = false;
  for (int dy = -1; dy <= 1; ++dy)
    for (int dx = -1; dx <= 1; ++dx) {
      int yy = y + dy, xx = x + dx;
      if (yy >= 0 && yy < h && xx >= 0 && xx < w && ed[(size_t)yy * w + xx] >= thr) dil = true;
    }
  float ei = dil ? 0.f : e_img[p];
  const bool border5 = (y < 5) || (y >= h - 5) || (x < 5) || (x >= w - 5);
  float en = (dil || border5) ? 0.f : e_nrm[p];
  e_img[p] = ei;
  e_nrm[p] = en;
  // hw % 256 == 0 so every block is within one image
  for (int off = 16; off; off >>= 1) {
    ei = fmaxf(ei, __shfl_xor(ei, off, 32));
    en = fmaxf(en, __shfl_xor(en, off, 32));
  }
  if ((threadIdx.x & 31) == 0) {
    atomic_max_pos(&maxes[im * 2 + 0], ei);
    atomic_max_pos(&maxes[im * 2 + 1], en);
  }
}

// ---------- K3: ordered compaction of edge pixels (stable-argsort equivalent) ----------

__global__ void k_compact(const float* __restrict__ e_img, const float* __restrict__ e_nrm,
                          const float* __restrict__ maxes, int* o_img, int* o_nrm,
                          int* counts, int n, int hw) {
  const int sid = blockIdx.x;
  const int im = sid >> 1, t = sid & 1;
  const float* e = (t ? e_nrm : e_img) + (size_t)im * hw;
  int* ord = (t ? o_nrm : o_img) + (size_t)im * hw;
  const float thr = 0.1f * maxes[sid];
  __shared__ unsigned wofs[32];
  __shared__ unsigned runbase;
  const int tid = threadIdx.x, lane = tid & 31, wid = tid >> 5;
  if (tid == 0) runbase = 0;
  __syncthreads();
  for (int base = 0; base < hw; base += (int)blockDim.x) {
    const int idx = base + tid;
    const bool p = (idx < hw) && (e[idx] >= thr);
    const unsigned b32 = (unsigned)__ballot(p);
    const unsigned prefix = __popc(b32 & ((1u << lane) - 1u));
    if (lane == 0) wofs[wid] = __popc(b32);
    __syncthreads();
    if (tid == 0) {
      unsigned s = runbase;
      for (int k = 0; k < (int)(blockDim.x >> 5); ++k) { unsigned c = wofs[k]; wofs[k] = s; s += c; }
      runbase = s;
    }
    __syncthreads();
    if (p) ord[wofs[wid] + prefix] = idx;
    __syncthreads();
  }
  if (tid == 0) counts[sid] = (int)runbase;
}

// ---------- K4: edge-guided sampling + pair-cosine loss, WMMA block reduction ----------

__global__ void k_sample(const float* __restrict__ inp, const float* __restrict__ tgt,
                         const float* __restrict__ t_img, const float* __restrict__ t_nrm,
                         const int* __restrict__ o_img, const int* __restrict__ o_nrm,
                         const int* __restrict__ counts, float* accum,
                         int n, int h, int w) {
  const int sid = blockIdx.y;
  const int im = sid >> 1, t = sid & 1;
  const size_t hw = (size_t)h * w;
  const float* thetas = (t ? t_nrm : t_img) + (size_t)im * hw;
  const int* ord = (t ? o_nrm : o_img) + (size_t)im * hw;
  int count = counts[sid]; if (count < 1) count = 1;
  const float* inpb = inp + (size_t)im * 3 * hw;
  const float* tgtb = tgt + (size_t)im * 3 * hw;
  const int s = blockIdx.x * blockDim.x + threadIdx.x;
  float partial = 0.f;
  if (s < S_PAIRS) {
    unsigned st = pcg_hash(0x2545F491u ^ (unsigned)(sid * 0x9E3779B9u) ^ (unsigned)(s * 0x85EBCA6Bu));
    st = pcg_hash(st);
    const int anchor = ord[st % (unsigned)count];
    const float th = thetas[anchor];
    float sn, cn;
    __sincosf(th, &sn, &cn);
    const float ac = fabsf(cn), asn = fabsf(sn);
    const int row_a = anchor / w, col_a = anchor % w;
    int flat[4];
    for (int k = 0; k < 4; ++k) {
      st = pcg_hash(st);
      float d = (float)(3u + st % 17u) * ((k < 2) ? -1.f : 1.f);
      int col = col_a + (int)rintf(d * ac);
      int row = row_a + (int)rintf(d * asn);
      col = col < 0 ? 0 : (col > w - 1 ? w - 1 : col);
      row = row < 0 ? 0 : (row > h - 1 ? h - 1 : row);
      flat[k] = row * w + col;
    }
    for (int q = 0; q < 3; ++q) {
      const int A = flat[q], B = flat[q + 1];
      float tc = 0.f, ic = 0.f;
      for (int c = 0; c < 3; ++c) {
        tc += tgtb[(size_t)c * hw + A] * tgtb[(size_t)c * hw + B];
        ic += inpb[(size_t)c * hw + A] * inpb[(size_t)c * hw + B];
      }
      partial += fabsf(fabsf(tc) - fabsf(ic));
    }
  }
  // Wave-level reduction via V_WMMA_F32_16X16X4_F32:
  // A = ones (16x4), B holds per-lane partials (4x16) -> D row0 = column sums.
  // Executed unconditionally: EXEC is all-ones here (full 256-thread block, reconverged).
  float wavesum;
#if __has_builtin(__builtin_amdgcn_wmma_f32_16x16x4_f32)
  {
    v2f a; a[0] = 1.f; a[1] = 1.f;
    v2f b; b[0] = partial; b[1] = 0.f;
    v8f c = {0.f, 0.f, 0.f, 0.f, 0.f, 0.f, 0.f, 0.f};
    c = __builtin_amdgcn_wmma_f32_16x16x4_f32(false, a, false, b, (short)0, c, false, false);
    float v = c[0];                 // 16 column sums live in lanes 0..15 (dup in 16..31 as row 8)
    v += __shfl_xor(v, 8, 32);      // half-wave reduce over the 16 columns
    v += __shfl_xor(v, 4, 32);
    v += __shfl_xor(v, 2, 32);
    v += __shfl_xor(v, 1, 32);
    wavesum = v;                    // lane 0 holds the wave total
  }
#else
  {
    float v = partial;
    for (int off = 16; off; off >>= 1) v += __shfl_xor(v, off, 32);
    wavesum = v;
  }
#endif
  __shared__ float wsum[8];
  const int lane = threadIdx.x & 31, wid = threadIdx.x >> 5;
  if (lane == 0) wsum[wid] = wavesum;
  __syncthreads();
  if (threadIdx.x == 0) {
    float tot = 0.f;
    for (int k = 0; k < (int)(blockDim.x >> 5); ++k) tot += wsum[k];
    atomicAdd(accum, tot);
  }
}

// ---------- K5: finalize mean ----------

__global__ void k_final(const float* __restrict__ accum, float* out, int denom) {
  if (threadIdx.x == 0 && blockIdx.x == 0) out[0] = accum[0] / (float)denom;
}

// ---------- launch ----------

extern "C" void kernel_launch(void* const* d_in, const int* in_sizes, int n_in,
                              void* d_out, int out_size, void* d_ws, size_t ws_size,
                              hipStream_t stream) {
  (void)n_in; (void)out_size; (void)ws_size;
  const float* gt  = (const float*)d_in[0];   // [n,1,h,w]
  const float* img = (const float*)d_in[1];   // [n,3,h,w]
  const float* inp = (const float*)d_in[2];   // [n,3,h,w]
  const float* tgt = (const float*)d_in[3];   // [n,3,h,w]
  const int h = H_IMG, w = W_IMG;
  const size_t hw = (size_t)h * w;
  const int n = (int)((size_t)in_sizes[0] / hw);

  // Workspace layout (needs 7*n*hw floats + 64 floats ~= 48 MB for n=8)
  float* ws = (float*)d_ws;
  float* e_img = ws + 0 * (size_t)n * hw;
  float* t_img = ws + 1 * (size_t)n * hw;
  float* e_nrm = ws + 2 * (size_t)n * hw;
  float* t_nrm = ws + 3 * (size_t)n * hw;
  float* e_dep = ws + 4 * (size_t)n * hw;
  int*   o_img = (int*)(ws + 5 * (size_t)n * hw);
  int*   o_nrm = (int*)(ws + 6 * (size_t)n * hw);
  float* smalls = ws + 7 * (size_t)n * hw;
  float* gmax   = smalls + 0;
  float* maxes  = smalls + 1;                 // 2n floats
  float* accum  = smalls + 1 + 2 * n;         // 1 float
  int*   counts = (int*)(smalls + 2 + 2 * n); // 2n ints

  k_init<<<1, 64, 0, stream>>>(smalls);

  dim3 b1(TILE_W, TILE_H);
  dim3 g1((w + TILE_W - 1) / TILE_W, n * ((h + TILE_H - 1) / TILE_H), 3);
  k_edges<<<g1, b1, 0, stream>>>(gt, img, tgt, e_img, t_img, e_nrm, t_nrm, e_dep, gmax, n, h, w);

  const int total = (int)((size_t)n * hw);
  k_mask<<<total / 256, 256, 0, stream>>>(e_img, e_nrm, e_dep, gmax, maxes, n, h, w);

  k_compact<<<2 * n, 1024, 0, stream>>>(e_img, e_nrm, maxes, o_img, o_nrm, counts, n, (int)hw);

  dim3 g4((S_PAIRS + 255) / 256, 2 * n);
  k_sample<<<g4, 256, 0, stream>>>(inp, tgt, t_img, t_nrm, o_img, o_nrm, counts, accum, n, h, w);

  k_final<<<1, 32, 0, stream>>>(accum, (float*)d_out, n * 6 * S_PAIRS);
}